// GCN_27049704030902
// MI455X (gfx1250) — compile-verified
//
#include <hip/hip_runtime.h>

typedef __attribute__((ext_vector_type(2))) float v2f;
typedef __attribute__((ext_vector_type(8))) float v8f;

// ---------------------------------------------------------------- utilities

__global__ void fill_zero_kernel(float* __restrict__ p, int n) {
    int i = blockIdx.x * blockDim.x + threadIdx.x;
    if (i < n) p[i] = 0.0f;
}

__global__ void degree_kernel(const int* __restrict__ col, float* __restrict__ deg, int E) {
    int i = blockIdx.x * blockDim.x + threadIdx.x;
    if (i < E) atomicAdd(&deg[col[i]], 1.0f);
}

__global__ void dinv_kernel(float* __restrict__ deg, int n) {
    int i = blockIdx.x * blockDim.x + threadIdx.x;
    if (i < n) deg[i] = rsqrtf(deg[i] + 1.0f);
}

// ------------------------------------------------- dense GEMM via fp32 WMMA
// H[n, cout] = X[n, cin] @ W[cin, cout], cout zero-padded to coutP = TN*16.
//
// Block = 256 threads = 8 waves. The block cooperatively stages W into LDS in
// a k-pair-interleaved layout:
//     lds[q*(2*coutP) + 2*c + (k&1)] = W[k][c],  q = k/2
// so each B fragment (the two K-values a lane needs for one N-tile) is a
// single contiguous 8-byte ds_load_b64 straight into an even register pair —
// no repacking movs before the WMMA. Zero-padding removes every divergent
// guard from the inner loop, so EXEC is all-ones around each WMMA.
//
// Each wave owns one 16-row M-tile and all TN N-tiles: one A fragment
// (global float2) feeds TN back-to-back v_wmma_f32_16x16x4_f32.
//
// Fragment layouts per CDNA5 ISA: A 16x4 f32 -> lanes 0-15 hold M=lane with
// K={0,1}, lanes 16-31 hold M=lane-16 with K={2,3}; B 4x16 mirrored over N;
// C/D 8 VGPRs with lanes 16-31 carrying M+8.
template <int TN>
__global__ void gemm_wmma_f32_kernel(const float* __restrict__ X,
                                     const float* __restrict__ W,
                                     float* __restrict__ H,
                                     int n, int cin, int cout) {
    constexpr int coutP = TN * 16;
    extern __shared__ float lds[];

    // ---- cooperative, padded, k-pair-interleaved staging of W ----
    const int total = cin * coutP;
    for (int i = threadIdx.x; i < total; i += blockDim.x) {
        const int k = i / coutP;
        const int c = i - k * coutP;
        const float v = (c < cout) ? W[(size_t)k * cout + c] : 0.0f;
        lds[(k >> 1) * (2 * coutP) + (c << 1) + (k & 1)] = v;
    }
    __syncthreads();

    const int lane   = threadIdx.x & 31;
    const int waveId = threadIdx.x >> 5;
    const int tileM  = blockIdx.x * (blockDim.x >> 5) + waveId;
    const int tilesM = (n + 15) >> 4;
    if (tileM >= tilesM) return;             // wave-uniform: EXEC stays full

    const int nlane = lane & 15;
    const int koff  = (lane >> 4) << 1;      // 0 or 2
    const int mrow  = (tileM << 4) + nlane;
    const int arow  = (mrow < n) ? mrow : (n - 1);

    const float* aPtr = X + (size_t)arow * cin + koff;
    // element index of this lane's B fragment for k-pair q = koff/2, col nlane
    int bIdx = (koff >> 1) * (2 * coutP) + (nlane << 1);

    v8f acc[TN] = {};
    for (int kb = 0; kb < cin; kb += 4) {
        const float2 av = *(const float2*)aPtr;
        v2f a; a.x = av.x; a.y = av.y;
        #pragma unroll
        for (int t = 0; t < TN; ++t) {
            const v2f b = *(const v2f*)(lds + bIdx + (t << 5));  // +t*16 cols *2
            acc[t] = __builtin_amdgcn_wmma_f32_16x16x4_f32(
                         /*neg_a=*/false, a, /*neg_b=*/false, b,
                         /*c_mod=*/(short)0, acc[t],
                         /*reuse_a=*/false, /*reuse_b=*/false);
        }
        aPtr += 4;
        bIdx += 4 * coutP;   // advance 2 k-pairs
    }

    // ---- store D tiles (mask only the padded columns, once) ----
    const int rbase = (tileM << 4) + ((lane >> 4) << 3);  // lanes>=16 -> M+8
    #pragma unroll
    for (int t = 0; t < TN; ++t) {
        const int col = t * 16 + nlane;
        if (col < cout) {
            #pragma unroll
            for (int r = 0; r < 8; ++r) {
                const int orow = rbase + r;
                if (orow < n) H[(size_t)orow * cout + col] = acc[t][r];
            }
        }
    }
}

// ----------------------------------------------------------- edge scatter
// agg[col[e], :] += h[row[e], :] * dinv[row[e]] * dinv[col[e]]
// cout/4 threads per edge, float4 gather (L2-resident), f32 global atomics.
__global__ void scatter_kernel(const float* __restrict__ h,
                               const int* __restrict__ rowIdx,
                               const int* __restrict__ colIdx,
                               const float* __restrict__ dinv,
                               float* __restrict__ agg,
                               int E, int cout) {
    const int per = cout >> 2;
    const int tid = blockIdx.x * blockDim.x + threadIdx.x;
    const int e   = tid / per;
    if (e >= E) return;
    const int c   = (tid - e * per) << 2;

    const int r = rowIdx[e];
    const int d = colIdx[e];
    const float nrm = dinv[r] * dinv[d];

    const float4 hv = *(const float4*)(h + (size_t)r * cout + c);
    float* dst = agg + (size_t)d * cout + c;
    atomicAdd(dst + 0, hv.x * nrm);
    atomicAdd(dst + 1, hv.y * nrm);
    atomicAdd(dst + 2, hv.z * nrm);
    atomicAdd(dst + 3, hv.w * nrm);
}

// ------------------------------------------------------------- finalize
// out = [relu](agg + h * dinv^2 + bias)
__global__ void finalize_kernel(const float* __restrict__ h,
                                const float* __restrict__ agg,
                                const float* __restrict__ dinv,
                                const float* __restrict__ bias,
                                float* __restrict__ out,
                                int n, int cout, int doRelu) {
    const int tid = blockIdx.x * blockDim.x + threadIdx.x;
    const int i   = tid / cout;
    if (i >= n) return;
    const int c   = tid - i * cout;
    const float s = dinv[i];
    float v = agg[(size_t)i * cout + c] + h[(size_t)i * cout + c] * (s * s) + bias[c];
    if (doRelu) v = fmaxf(v, 0.0f);
    out[(size_t)i * cout + c] = v;
}

// ----------------------------------------------------------------- driver

static inline int cdiv(long long a, long long b) { return (int)((a + b - 1) / b); }

static void launch_gemm(const float* X, const float* W, float* H,
                        int n, int cin, int cout, hipStream_t stream) {
    const int tilesM = (n + 15) >> 4;
    const int blocks = cdiv(tilesM, 8);       // 8 waves (M-tiles) per block
    const int TN     = (cout + 15) >> 4;
    const size_t shmem = (size_t)cin * TN * 16 * sizeof(float);
    switch (TN) {
        case 4: gemm_wmma_f32_kernel<4><<<blocks, 256, shmem, stream>>>(X, W, H, n, cin, cout); break;
        case 3: gemm_wmma_f32_kernel<3><<<blocks, 256, shmem, stream>>>(X, W, H, n, cin, cout); break;
        case 2: gemm_wmma_f32_kernel<2><<<blocks, 256, shmem, stream>>>(X, W, H, n, cin, cout); break;
        default: gemm_wmma_f32_kernel<1><<<blocks, 256, shmem, stream>>>(X, W, H, n, cin, cout); break;
    }
}

extern "C" void kernel_launch(void* const* d_in, const int* in_sizes, int n_in,
                              void* d_out, int out_size, void* d_ws, size_t ws_size,
                              hipStream_t stream) {
    const float* x    = (const float*)d_in[0];
    const int*   ei   = (const int*)  d_in[1];
    const float* W1   = (const float*)d_in[2];
    const float* b1   = (const float*)d_in[3];
    const float* W2   = (const float*)d_in[4];
    const float* b2   = (const float*)d_in[5];
    const float* Wout = (const float*)d_in[6];
    const float* bout = (const float*)d_in[7];

    const int C1  = in_sizes[3];            // 64
    const int C2  = in_sizes[5];            // 64
    const int C3  = in_sizes[7];            // 40
    const int Cin = in_sizes[2] / C1;       // 128
    const int n   = in_sizes[0] / Cin;      // 100000
    const int E   = in_sizes[1] / 2;        // 1600000

    const int* rowIdx = ei;
    const int* colIdx = ei + E;

    // workspace: dinv[n] | hbuf[n*64] | abuf[n*64]
    float* dinv = (float*)d_ws;
    float* hbuf = dinv + n;
    float* abuf = hbuf + (size_t)n * C1;

    const int T = 256;

    // --- degree -> dinv (shared by all three layers) ---
    fill_zero_kernel<<<cdiv(n, T), T, 0, stream>>>(dinv, n);
    degree_kernel<<<cdiv(E, T), T, 0, stream>>>(colIdx, dinv, E);
    dinv_kernel<<<cdiv(n, T), T, 0, stream>>>(dinv, n);

    // ---------------- layer 1: x[n,Cin] -> abuf[n,C1], relu ----------------
    launch_gemm(x, W1, hbuf, n, Cin, C1, stream);
    fill_zero_kernel<<<cdiv((long long)n * C1, T), T, 0, stream>>>(abuf, n * C1);
    scatter_kernel<<<cdiv((long long)E * (C1 >> 2), T), T, 0, stream>>>(hbuf, rowIdx, colIdx, dinv, abuf, E, C1);
    finalize_kernel<<<cdiv((long long)n * C1, T), T, 0, stream>>>(hbuf, abuf, dinv, b1, abuf, n, C1, 1);

    // ---------------- layer 2: abuf[n,C1] -> abuf[n,C2], relu --------------
    launch_gemm(abuf, W2, hbuf, n, C1, C2, stream);
    fill_zero_kernel<<<cdiv((long long)n * C2, T), T, 0, stream>>>(abuf, n * C2);
    scatter_kernel<<<cdiv((long long)E * (C2 >> 2), T), T, 0, stream>>>(hbuf, rowIdx, colIdx, dinv, abuf, E, C2);
    finalize_kernel<<<cdiv((long long)n * C2, T), T, 0, stream>>>(hbuf, abuf, dinv, b2, abuf, n, C2, 1);

    // ---------------- layer 3: abuf[n,C2] -> d_out[n,C3], no relu ----------
    launch_gemm(abuf, Wout, hbuf, n, C2, C3, stream);
    fill_zero_kernel<<<cdiv((long long)n * C3, T), T, 0, stream>>>(abuf, n * C3);
    scatter_kernel<<<cdiv((long long)E * (C3 >> 2), T), T, 0, stream>>>(hbuf, rowIdx, colIdx, dinv, abuf, E, C3);
    finalize_kernel<<<cdiv((long long)n * C3, T), T, 0, stream>>>(hbuf, abuf, dinv, bout, (float*)d_out, n, C3, 0);
}